// JODIE_10307921510829
// MI455X (gfx1250) — compile-verified
//
#include <hip/hip_runtime.h>
#include <math.h>

// ---- problem constants (match reference) ----
#define NB   4096     // batch B
#define DM   128      // d_model D
#define K4   512      // 4*D (pred GEMM K)
#define N2   256      // 2*D (pred GEMM N)
#define NUSR 200000
#define NITM 200000

typedef float v2f __attribute__((ext_vector_type(2)));
typedef float v8f __attribute__((ext_vector_type(8)));

// -------------------------------------------------------------------------
// Gather + elementwise + pack the 4*D-wide GEMM input X into workspace.
// One block per batch row, 128 threads (one per feature).
// -------------------------------------------------------------------------
__global__ void jodie_gather_pack(
    const int* __restrict__ user_id, const int* __restrict__ prev_item_id,
    const int* __restrict__ item_id, const float* __restrict__ t_item,
    const float* __restrict__ dyn_user, const float* __restrict__ dyn_item,
    const float* __restrict__ is_user_new, const float* __restrict__ is_item_new,
    const float* __restrict__ static_user_W, const float* __restrict__ static_item_W,
    const float* __restrict__ init_user, const float* __restrict__ init_item,
    const float* __restrict__ td_W, const float* __restrict__ td_b,
    float* __restrict__ X,              // (B, 512) workspace
    float* __restrict__ out_user_emb,   // (B, 128)
    float* __restrict__ out_item_emb,   // (B, 128)
    float* __restrict__ out_target)     // (B, 256)
{
    const int b = blockIdx.x;
    const int d = threadIdx.x;
    const int u  = user_id[b];
    const int it = item_id[b];
    const int pi = prev_item_id[b];

    const float ue   = is_user_new[u]  * init_user[d] + dyn_user[(size_t)u  * DM + d];
    const float ie   = is_item_new[it] * init_item[d] + dyn_item[(size_t)it * DM + d];
    const float pie  = is_item_new[pi] * init_item[d] + dyn_item[(size_t)pi * DM + d];
    const float sue  = static_user_W[(size_t)u  * DM + d];
    const float sie  = static_item_W[(size_t)it * DM + d];
    const float psie = static_item_W[(size_t)pi * DM + d];

    const float proj = 1.0f + t_item[b] * td_W[d] + td_b[d];

    out_user_emb[b * DM + d] = ue;
    out_item_emb[b * DM + d] = ie;
    out_target[b * N2 + d]        = ie;
    out_target[b * N2 + DM + d]   = sie;

    float* xb = X + (size_t)b * K4;
    xb[d]           = ue * proj;   // user_proj
    xb[DM + d]      = pie;         // prev_item_emb
    xb[2 * DM + d]  = psie;        // prev_static_item_emb
    xb[3 * DM + d]  = sue;         // static_user_emb
}

// -------------------------------------------------------------------------
// item_pred = X (4096x512) @ pred_W^T (512x256) + pred_b, via f32 WMMA.
// One wave per 16x16 output tile; K-loop in steps of 4.
// A lane layout (16x4 f32): M = lane%16, K = 2*(lane>=16) + {0,1}
// B lane layout (4x16 f32): N = lane%16, K = 2*(lane>=16) + {0,1} (B = W^T)
// D lane layout (16x16 f32): VGPR v -> M = v + 8*(lane>=16), N = lane%16
// -------------------------------------------------------------------------
__global__ void jodie_pred_gemm(
    const float* __restrict__ X,      // (4096, 512)
    const float* __restrict__ W,      // (256, 512) row-major (pred_W)
    const float* __restrict__ bias,   // (256,)
    float* __restrict__ out)          // (4096, 256)
{
    const int wave = (blockIdx.x * blockDim.x + threadIdx.x) >> 5;
    const int lane = threadIdx.x & 31;
    const int half = lane >> 4;
    const int l16  = lane & 15;
    const int nt = wave & 15;   // 256/16 n-tiles
    const int mt = wave >> 4;   // 4096/16 m-tiles

    const float* arow = X + (size_t)(mt * 16 + l16) * K4 + 2 * half;
    const float* brow = W + (size_t)(nt * 16 + l16) * K4 + 2 * half;

    v8f acc = {};
    #pragma unroll 4
    for (int k = 0; k < K4; k += 4) {
        v2f a, b;
        a.x = arow[k]; a.y = arow[k + 1];
        b.x = brow[k]; b.y = brow[k + 1];
        acc = __builtin_amdgcn_wmma_f32_16x16x4_f32(
            false, a, false, b, (short)0, acc, false, false);
    }

    const int n = nt * 16 + l16;
    const float bn = bias[n];
    #pragma unroll
    for (int v = 0; v < 8; ++v) {
        const int m = mt * 16 + v + 8 * half;
        out[(size_t)m * N2 + n] = acc[v] + bn;
    }
}

// -------------------------------------------------------------------------
// RNNCell: out = tanh(Xemb @ Wih[:, :128]^T + t*Wih[:,128] + bih
//                     + H @ Whh^T + bhh)
// Both K=128 products fused into one WMMA accumulator per 16x16 tile.
// Wih is (128, 129) row-major -> scalar-pair loads (odd rows 4B-misaligned).
// -------------------------------------------------------------------------
__global__ void jodie_rnn_gemm(
    const float* __restrict__ Xemb,   // (4096, 128) input-side embedding
    const float* __restrict__ H,      // (4096, 128) hidden-side embedding
    const float* __restrict__ Wih,    // (128, 129)
    const float* __restrict__ Whh,    // (128, 128)
    const float* __restrict__ bih, const float* __restrict__ bhh,
    const float* __restrict__ t,      // (4096,)
    float* __restrict__ out)          // (4096, 128)
{
    const int wave = (blockIdx.x * blockDim.x + threadIdx.x) >> 5;
    const int lane = threadIdx.x & 31;
    const int half = lane >> 4;
    const int l16  = lane & 15;
    const int nt = wave & 7;    // 128/16 n-tiles
    const int mt = wave >> 3;   // 4096/16 m-tiles
    const int n = nt * 16 + l16;

    const float* a1 = Xemb + (size_t)(mt * 16 + l16) * DM + 2 * half;
    const float* a2 = H    + (size_t)(mt * 16 + l16) * DM + 2 * half;
    const float* b1 = Wih  + (size_t)n * (DM + 1) + 2 * half;
    const float* b2 = Whh  + (size_t)n * DM + 2 * half;

    v8f acc = {};
    #pragma unroll 2
    for (int k = 0; k < DM; k += 4) {
        v2f av, bv;
        av.x = a1[k]; av.y = a1[k + 1];
        bv.x = b1[k]; bv.y = b1[k + 1];
        acc = __builtin_amdgcn_wmma_f32_16x16x4_f32(
            false, av, false, bv, (short)0, acc, false, false);
        av.x = a2[k]; av.y = a2[k + 1];
        bv.x = b2[k]; bv.y = b2[k + 1];
        acc = __builtin_amdgcn_wmma_f32_16x16x4_f32(
            false, av, false, bv, (short)0, acc, false, false);
    }

    const float bn = bih[n] + bhh[n];
    const float wt = Wih[(size_t)n * (DM + 1) + DM];   // time column
    #pragma unroll
    for (int v = 0; v < 8; ++v) {
        const int m = mt * 16 + v + 8 * half;
        out[(size_t)m * DM + n] = tanhf(acc[v] + bn + t[m] * wt);
    }
}

// -------------------------------------------------------------------------
// Scatter updated embeddings into the (already memcpy'd) persistent buffers
// and clear the is_new flags at touched rows.
// -------------------------------------------------------------------------
__global__ void jodie_scatter(
    const int* __restrict__ user_id, const int* __restrict__ item_id,
    const float* __restrict__ upd_user, const float* __restrict__ upd_item,
    float* __restrict__ new_dyn_user, float* __restrict__ new_dyn_item,
    float* __restrict__ new_is_user, float* __restrict__ new_is_item)
{
    const int b = blockIdx.x;
    const int d = threadIdx.x;
    const int u  = user_id[b];
    const int it = item_id[b];
    new_dyn_user[(size_t)u  * DM + d] = upd_user[b * DM + d];
    new_dyn_item[(size_t)it * DM + d] = upd_item[b * DM + d];
    if (d == 0) {
        new_is_user[u]  = 0.0f;
        new_is_item[it] = 0.0f;
    }
}

extern "C" void kernel_launch(void* const* d_in, const int* in_sizes, int n_in,
                              void* d_out, int out_size, void* d_ws, size_t ws_size,
                              hipStream_t stream) {
    // ---- inputs (setup_inputs dict order) ----
    const int*   user_id        = (const int*)  d_in[0];
    const int*   prev_item_id   = (const int*)  d_in[1];
    const int*   item_id        = (const int*)  d_in[2];
    const float* t_item         = (const float*)d_in[3];
    const float* t_user         = (const float*)d_in[4];
    const float* dyn_user       = (const float*)d_in[5];
    const float* dyn_item       = (const float*)d_in[6];
    const float* is_user_new    = (const float*)d_in[7];
    const float* is_item_new    = (const float*)d_in[8];
    const float* static_user_W  = (const float*)d_in[9];
    const float* static_item_W  = (const float*)d_in[10];
    const float* init_user      = (const float*)d_in[11];
    const float* init_item      = (const float*)d_in[12];
    const float* user_Wih       = (const float*)d_in[13];
    const float* user_Whh       = (const float*)d_in[14];
    const float* user_bih       = (const float*)d_in[15];
    const float* user_bhh       = (const float*)d_in[16];
    const float* item_Wih       = (const float*)d_in[17];
    const float* item_Whh       = (const float*)d_in[18];
    const float* item_bih       = (const float*)d_in[19];
    const float* item_bhh       = (const float*)d_in[20];
    const float* pred_W         = (const float*)d_in[21];
    const float* pred_b         = (const float*)d_in[22];
    const float* td_W           = (const float*)d_in[23];
    const float* td_b           = (const float*)d_in[24];

    // ---- outputs (flat, return order) ----
    float* out = (float*)d_out;
    float* o_pred      = out;                                  // (B, 256)
    float* o_target    = o_pred   + (size_t)NB * N2;           // (B, 256)
    float* o_upd_user  = o_target + (size_t)NB * N2;           // (B, 128)
    float* o_user_emb  = o_upd_user + (size_t)NB * DM;         // (B, 128)
    float* o_upd_item  = o_user_emb + (size_t)NB * DM;         // (B, 128)
    float* o_item_emb  = o_upd_item + (size_t)NB * DM;         // (B, 128)
    float* o_dyn_user  = o_item_emb + (size_t)NB * DM;         // (NU, 128)
    float* o_dyn_item  = o_dyn_user + (size_t)NUSR * DM;       // (NI, 128)
    float* o_is_user   = o_dyn_item + (size_t)NITM * DM;       // (NU,)
    float* o_is_item   = o_is_user  + (size_t)NUSR;            // (NI,)

    float* X = (float*)d_ws;                                   // (B, 512)

    // 1) bulk D2D copies of the persistent buffers (memory-bound part,
    //    ~410 MB total traffic -> dominates runtime at 23.3 TB/s)
    hipMemcpyAsync(o_dyn_user, dyn_user, (size_t)NUSR * DM * sizeof(float),
                   hipMemcpyDeviceToDevice, stream);
    hipMemcpyAsync(o_dyn_item, dyn_item, (size_t)NITM * DM * sizeof(float),
                   hipMemcpyDeviceToDevice, stream);
    hipMemcpyAsync(o_is_user, is_user_new, (size_t)NUSR * sizeof(float),
                   hipMemcpyDeviceToDevice, stream);
    hipMemcpyAsync(o_is_item, is_item_new, (size_t)NITM * sizeof(float),
                   hipMemcpyDeviceToDevice, stream);

    // 2) gathers + elementwise + pack GEMM input
    jodie_gather_pack<<<NB, DM, 0, stream>>>(
        user_id, prev_item_id, item_id, t_item,
        dyn_user, dyn_item, is_user_new, is_item_new,
        static_user_W, static_item_W, init_user, init_item,
        td_W, td_b, X, o_user_emb, o_item_emb, o_target);

    // 3) item_pred GEMM: 4096x512x256 via v_wmma_f32_16x16x4_f32
    //    (4096/16)*(256/16) = 4096 tiles; 8 waves/block -> 512 blocks
    jodie_pred_gemm<<<512, 256, 0, stream>>>(X, pred_W, pred_b, o_pred);

    // 4) RNN cells: 2048 tiles each; 8 waves/block -> 256 blocks
    //    user cell: x = [item_emb, t_item], h = user_emb
    jodie_rnn_gemm<<<256, 256, 0, stream>>>(
        o_item_emb, o_user_emb, user_Wih, user_Whh, user_bih, user_bhh,
        t_item, o_upd_user);
    //    item cell: x = [user_emb, t_user], h = item_emb
    jodie_rnn_gemm<<<256, 256, 0, stream>>>(
        o_user_emb, o_item_emb, item_Wih, item_Whh, item_bih, item_bhh,
        t_user, o_upd_item);

    // 5) scatter updated rows + clear flags (stream-ordered after memcpys)
    jodie_scatter<<<NB, DM, 0, stream>>>(
        user_id, item_id, o_upd_user, o_upd_item,
        o_dyn_user, o_dyn_item, o_is_user, o_is_item);
}